// LocalConnect_62285615726786
// MI455X (gfx1250) — compile-verified
//
#include <hip/hip_runtime.h>

// LocalConnect: out[b,o,x,y] = sum_{c,k} patch[b,c,x,y,k] * w[o,c,x,y,k]
// B=32, IN_C=64, OUT_C=64, IMG=OH=32, K=3, S=1, P=1.
// Per spatial position: GEMM M=32, N=64, K=576. 1024 independent positions.
// Memory-bound (151MB weights @ 23.3TB/s ~= 7us floor) -> fp32 WMMA, weights
// streamed through LDS exactly once.

typedef __attribute__((ext_vector_type(2))) float v2f;
typedef __attribute__((ext_vector_type(8))) float v8f;

namespace lc {
constexpr int BATCH = 32;
constexpr int INC   = 64;
constexpr int OUTC  = 64;
constexpr int IMG   = 32;
constexpr int KS    = 3;
constexpr int OH    = 32;
constexpr int KDIM  = INC * KS * KS;   // 576
constexpr int CHUNK = 144;             // 16 input channels * 9 taps
constexpr int NCHUNK = KDIM / CHUNK;   // 4
constexpr int LDSTR  = CHUNK + 1;      // 145 (odd -> LDS bank-conflict free)

// strides
constexpr int X_B = INC * IMG * IMG;       // 65536
constexpr int X_C = IMG * IMG;             // 1024
constexpr int W_O = INC * OH * OH * KS*KS; // 589824
constexpr int W_C = OH * OH * KS * KS;     // 9216
constexpr int W_X = OH * KS * KS;          // 288
constexpr int W_Y = KS * KS;               // 9
constexpr int O_B = OUTC * OH * OH;        // 65536
constexpr int O_O = OH * OH;               // 1024
} // namespace lc

__global__ __launch_bounds__(256)
void LocalConnect_62285615726786_kernel(const float* __restrict__ x,
                                        const float* __restrict__ w,
                                        float* __restrict__ out) {
    using namespace lc;

    __shared__ float As[BATCH][LDSTR];  // patches tile: 32 x 144 (+pad)
    __shared__ float Bs[OUTC][LDSTR];   // weight tile:  64 x 144 (+pad)

    const int pos = blockIdx.x;         // 0..1023
    const int ox  = pos >> 5;
    const int oy  = pos & 31;

    const int tid  = threadIdx.x;       // 0..255 : 8 wave32s
    const int wave = tid >> 5;
    const int lane = tid & 31;
    const int row  = lane & 15;         // M (A) or N (B/C) index within tile
    const int hi   = lane >> 4;         // lane group 0/1 -> K offset 0/2

    const int mt = wave >> 2;           // batch tile   (0..1)
    const int nt = wave & 3;            // channel tile (0..3)

    const int wpos = ox * W_X + oy * W_Y;   // per-position weight offset

    v8f acc = {};                        // 16x16 f32 accumulator (8 VGPRs)

    for (int ch = 0; ch < NCHUNK; ++ch) {
        const int c_base = ch * (CHUNK / (KS * KS));   // 16 channels per chunk
        __syncthreads();                                // LDS reuse from prev chunk

        // --- cooperative stage: patches (32x144) ---
        // contiguous over the 9-tap runs of x rows; zero-fill at borders
        for (int t = tid; t < BATCH * CHUNK; t += 256) {
            const int m  = t / CHUNK;
            const int kc = t % CHUNK;
            const int c  = c_base + kc / (KS * KS);
            const int k  = kc % (KS * KS);
            const int ix = ox - 1 + k / KS;
            const int iy = oy - 1 + k % KS;
            float v = 0.0f;
            if ((unsigned)ix < (unsigned)IMG && (unsigned)iy < (unsigned)IMG)
                v = x[m * X_B + c * X_C + ix * IMG + iy];
            As[m][kc] = v;
        }
        // --- cooperative stage: weights (64x144), read from HBM exactly once ---
        for (int t = tid; t < OUTC * CHUNK; t += 256) {
            const int o  = t / CHUNK;
            const int kc = t % CHUNK;
            const int c  = c_base + kc / (KS * KS);
            const int k  = kc % (KS * KS);
            Bs[o][kc] = w[o * W_O + c * W_C + wpos + k];
        }
        __syncthreads();

        // --- compute: 36 x V_WMMA_F32_16X16X4_F32 per wave per chunk ---
        const float* arow = &As[mt * 16 + row][0];
        const float* brow = &Bs[nt * 16 + row][0];
        #pragma unroll
        for (int ks = 0; ks < CHUNK; ks += 4) {
            const int kA = ks + 2 * hi;          // ISA A/B f32 fragment layout:
            v2f a, b;                            // VGPR0 = K {0|2}, VGPR1 = K {1|3}
            a.x = arow[kA];
            a.y = arow[kA + 1];
            b.x = brow[kA];
            b.y = brow[kA + 1];
            acc = __builtin_amdgcn_wmma_f32_16x16x4_f32(
                /*neg_a=*/false, a, /*neg_b=*/false, b,
                /*c_mod=*/(short)0, acc, /*reuse_a=*/false, /*reuse_b=*/false);
        }
    }

    // --- store: C/D layout -> lane holds N=row, VGPR v holds M = v + 8*hi ---
    const int o = nt * 16 + row;
    #pragma unroll
    for (int v = 0; v < 8; ++v) {
        const int b = mt * 16 + v + 8 * hi;
        out[b * O_B + o * O_O + pos] = acc[v];
    }
}

extern "C" void kernel_launch(void* const* d_in, const int* in_sizes, int n_in,
                              void* d_out, int out_size, void* d_ws, size_t ws_size,
                              hipStream_t stream) {
    const float* x = (const float*)d_in[0];
    const float* w = (const float*)d_in[1];
    float* out     = (float*)d_out;
    (void)in_sizes; (void)n_in; (void)out_size; (void)d_ws; (void)ws_size;

    dim3 grid(lc::OH * lc::OH);   // 1024 positions
    dim3 block(256);              // 8 wave32s -> 8 WMMA tiles = full 32x64 output
    LocalConnect_62285615726786_kernel<<<grid, block, 0, stream>>>(x, w, out);
}